// ParcelRebalancedLDAM_13477607375414
// MI455X (gfx1250) — compile-verified
//
#include <hip/hip_runtime.h>

// ---------------------------------------------------------------------------
// ParcelRebalancedLDAM for MI455X (gfx1250)
//  Phase 1: streaming scatter-reduce. Async b128 global->LDS staging (NT
//           cache policy), LDS transpose, then channel-contiguous atomics so
//           each wave-atomic touches 2 parcel rows (2 cache lines), not 32.
//  Phase 2: per-parcel LDAM log-softmax NLL + mean (tiny).
// ---------------------------------------------------------------------------

#define PARCELS   100000
#define NCH       16
#define HW_       (512 * 512)        // pixels per image plane
#define NPIX      (8 * HW_)          // total pixels M = 2,097,152
#define IGNORE_T  255
#define LDAM_S    30.0f
#define LDAM_MAXM 0.5f
#define TPAD      260                // tile row stride: 16B-aligned, bank-phase 4

// low 32 bits of a generic pointer to __shared__ data == LDS byte offset
__device__ __forceinline__ unsigned lds_off(const void* p) {
    return (unsigned)(unsigned long long)p;
}

// ---------------------------------------------------------------------------
// Workspace init (re-run every launch: deterministic, harness poisons d_ws)
// ---------------------------------------------------------------------------
__global__ void init_kernel(int* __restrict__ counts,
                            int* __restrict__ firstv,
                            float* __restrict__ sums,
                            float* __restrict__ acc) {
    const int total = 2 * PARCELS + NCH * PARCELS + 2;
    for (int i = blockIdx.x * blockDim.x + threadIdx.x; i < total;
         i += gridDim.x * blockDim.x) {
        if (i < PARCELS)                    counts[i] = 0;
        else if (i < 2 * PARCELS)           firstv[i - PARCELS] = 0x7FFFFFFF;
        else if (i < (2 + NCH) * PARCELS)   sums[i - 2 * PARCELS] = 0.0f;
        else                                acc[i - (2 + NCH) * PARCELS] = 0.0f;
    }
}

// ---------------------------------------------------------------------------
// Phase 1 scatter: 256 pixels per block.
// ---------------------------------------------------------------------------
__global__ void __launch_bounds__(256)
scatter_kernel(const float* __restrict__ pred,
               const int*   __restrict__ target,
               const int*   __restrict__ parcel,
               int*   __restrict__ counts,
               int*   __restrict__ firstv,
               float* __restrict__ sums) {
    __shared__ float tile[NCH][TPAD];   // staged pred tile, row-padded
    __shared__ int   pl[256];           // parcel id per pixel (-1 = ignore)

    const int tid = threadIdx.x;
    const int m0  = blockIdx.x * 256;             // HW_ % 256 == 0: one image
    const int n   = m0 / HW_;
    const int r0  = m0 - n * HW_;
    const float* gbase = pred + (size_t)n * (NCH * HW_) + r0;

    // --- A: async-stage 16KB tile, 4 x b128 wave-copies per wave, NT policy
    //        (don't let the 128MB read-once stream evict the L2 parcel table)
#pragma unroll
    for (int k = 0; k < 4; ++k) {
        const int chunk = k * 256 + tid;          // 1024 chunks of 4 pixels
        const int c = chunk >> 6;                 // channel
        const int q = (chunk & 63) << 2;          // first pixel of 4
        const float* g = gbase + c * HW_ + q;     // 16B, coalesced per wave
        unsigned la = lds_off(&tile[c][q]);       // 16B-aligned (TPAD*4%16==0)
        asm volatile("global_load_async_to_lds_b128 %0, %1, off th:TH_LOAD_NT"
                     :: "v"(la), "v"(g)
                     : "memory");
    }

    // --- B: overlapped control work: per-pixel counts/first + parcel to LDS
    const int m = m0 + tid;
    const int t = target[m];
    const int p = parcel[m];
    const bool valid = (t != IGNORE_T);
    pl[tid] = valid ? p : -1;
    if (valid) {
        atomicAdd(&counts[p], 1);
        atomicMin(&firstv[p], m);
    }

    asm volatile("s_wait_asynccnt 0x0" ::: "memory");
    __syncthreads();

    // --- C: transposed scatter. lane -> (channel c, pixel-slot s); each wave
    //        atomic covers 16 channels x 2 pixels = 2 contiguous 64B rows.
    const int c = tid & 15;
    const int s = tid >> 4;
#pragma unroll
    for (int j = 0; j < 16; ++j) {
        const int q  = s + (j << 4);
        const int pv = pl[q];                     // LDS broadcast read
        if (pv >= 0)
            atomicAdd(&sums[(size_t)pv * NCH + c], tile[c][q]); // conflict-free
    }
}

// ---------------------------------------------------------------------------
// Phase 2: one thread per parcel. Margins from cls_num_list, masked
// log-softmax NLL, wave32 shuffle reduction, one atomic per wave.
// ---------------------------------------------------------------------------
__global__ void __launch_bounds__(256)
finalize_kernel(const int*   __restrict__ target,
                const int*   __restrict__ cls_num,
                const int*   __restrict__ counts,
                const int*   __restrict__ firstv,
                const float* __restrict__ sums,
                float* __restrict__ acc) {
    const int p = blockIdx.x * blockDim.x + threadIdx.x;
    float nll = 0.0f, vflag = 0.0f;

    if (p < PARCELS) {
        const int cnt = counts[p];
        if (cnt > 0) {
            // m_j = max_m * n_j^{-1/4} / max_k n_k^{-1/4}
            float mlist[NCH];
            float mmax = 0.0f;
#pragma unroll
            for (int j = 0; j < NCH; ++j) {
                const float nj = (float)cls_num[j];
                const float v  = rsqrtf(sqrtf(nj));   // n^{-1/4}
                mlist[j] = v;
                mmax = fmaxf(mmax, v);
            }
            const float mscale = LDAM_MAXM / mmax;

            int f = firstv[p];
            if (f > NPIX - 1) f = NPIX - 1;
            const int tp = target[f];                 // true class of parcel

            const float inv = 1.0f / (float)cnt;
            const float* srow = sums + (size_t)p * NCH;
            float x[NCH];
            float xmax = -3.4e38f;
#pragma unroll
            for (int cch = 0; cch < NCH; ++cch) {
                float a = srow[cch] * inv;            // segment mean
                if (cch == tp) a -= mlist[tp] * mscale;
                x[cch] = LDAM_S * a;
                xmax = fmaxf(xmax, x[cch]);
            }
            float se = 0.0f;
#pragma unroll
            for (int cch = 0; cch < NCH; ++cch) se += expf(x[cch] - xmax);
            const float lse = xmax + logf(se);
            nll   = lse - x[tp];                      // -log_softmax at tp
            vflag = 1.0f;
        }
    }

    // wave32 reduction, one atomic per wave
#pragma unroll
    for (int o = 16; o > 0; o >>= 1) {
        nll   += __shfl_down(nll, o, 32);
        vflag += __shfl_down(vflag, o, 32);
    }
    if ((threadIdx.x & 31) == 0) {
        atomicAdd(&acc[0], nll);
        atomicAdd(&acc[1], vflag);
    }
}

__global__ void final_div_kernel(const float* __restrict__ acc,
                                 float* __restrict__ out) {
    if (threadIdx.x == 0 && blockIdx.x == 0)
        out[0] = acc[0] / fmaxf(acc[1], 1.0f);
}

// ---------------------------------------------------------------------------
extern "C" void kernel_launch(void* const* d_in, const int* in_sizes, int n_in,
                              void* d_out, int out_size, void* d_ws, size_t ws_size,
                              hipStream_t stream) {
    const float* pred    = (const float*)d_in[0];   // (8,16,512,512) f32
    const int*   target  = (const int*)d_in[1];     // (8,512,512) i32
    const int*   parcel  = (const int*)d_in[2];     // (8,512,512) i32
    const int*   cls_num = (const int*)d_in[3];     // (16,) i32

    // workspace layout: counts[P] | first[P] | sums[P*16] | acc[2]   (~7.2 MB)
    int*   counts = (int*)d_ws;
    int*   firstv = counts + PARCELS;
    float* sums   = (float*)(firstv + PARCELS);
    float* acc    = sums + (size_t)PARCELS * NCH;
    float* out    = (float*)d_out;

    init_kernel<<<1024, 256, 0, stream>>>(counts, firstv, sums, acc);
    scatter_kernel<<<NPIX / 256, 256, 0, stream>>>(pred, target, parcel,
                                                   counts, firstv, sums);
    finalize_kernel<<<(PARCELS + 255) / 256, 256, 0, stream>>>(target, cls_num,
                                                               counts, firstv,
                                                               sums, acc);
    final_div_kernel<<<1, 32, 0, stream>>>(acc, out);
}